// IDWT_14989435863554
// MI455X (gfx1250) — compile-verified
//
#include <hip/hip_runtime.h>
#include <stdint.h>

// idwt2 (Haar) : coeffs (4, 4096, 4096) f32 -> out (8192, 8192) f32
// Pure streaming problem: 512 MB total traffic, ~0.4 flop/byte -> HBM bound
// (~22 us at 23.3 TB/s). Strategy: TDM (tensor_load_to_lds) DMA staging of
// coefficient tiles into LDS with double buffering, VALU butterfly, and
// non-temporal b128 stores (no reuse of output; 256 MB > 192 MB L2).

#define CH 4096                 // coeff rows
#define CW 4096                 // coeff cols
#define OW 8192                 // output width
#define ROWS_PER_BLK 2          // coeff rows per block
#define CHUNK_W 1024            // coeff cols per chunk
#define NCHUNK (CW / CHUNK_W)   // 4
#define THREADS 256

typedef float        v4f   __attribute__((ext_vector_type(4)));
typedef unsigned int u32x4 __attribute__((ext_vector_type(4)));
typedef int          i32x4 __attribute__((ext_vector_type(4)));
typedef int          i32x8 __attribute__((ext_vector_type(8)));

#if defined(__gfx1250__) && __has_builtin(__builtin_amdgcn_tensor_load_to_lds) && \
    __has_builtin(__builtin_amdgcn_s_wait_tensorcnt)
#define USE_TDM 1
#else
#define USE_TDM 0
#endif

#if USE_TDM
// Issue one TDM 2-D tile load: ROWS_PER_BLK x CHUNK_W f32 tile from a 4096x4096
// plane into LDS at lds_byte_addr. Descriptor packed per CDNA5 ISA Ch.8 (D#).
__device__ __forceinline__ void tdm_load_tile(const float* gptr, uint32_t lds_byte_addr) {
  const uint64_t ga = (uint64_t)(uintptr_t)gptr;

  u32x4 g0;
  g0[0] = 1u;                                               // count=1, user mode
  g0[1] = lds_byte_addr;                                    // lds_addr [63:32]
  g0[2] = (uint32_t)ga;                                     // global_addr[31:0]
  g0[3] = (uint32_t)((ga >> 32) & 0x1FFFFFFu) | (2u << 30); // addr[56:32] | type=2

  i32x8 g1;
  g1[0] = (int)(2u << 16);                                  // data_size=2 (4B), mask=0
  g1[1] = (int)(((uint32_t)CW & 0xFFFFu) << 16);            // tensor_dim0[15:0] @ bits63:48
  g1[2] = (int)((((uint32_t)CW >> 16) & 0xFFFFu) |          // tensor_dim0[31:16]
                (((uint32_t)CH & 0xFFFFu) << 16));          // tensor_dim1[15:0]
  g1[3] = (int)((((uint32_t)CH >> 16) & 0xFFFFu) |          // tensor_dim1[31:16]
                ((uint32_t)CHUNK_W << 16));                 // tile_dim0
  g1[4] = (int)ROWS_PER_BLK;                                // tile_dim1 | tile_dim2=0
  g1[5] = (int)CW;                                          // tensor_dim0_stride[31:0]
  g1[6] = 0;                                                // stride0 hi | stride1 lo
  g1[7] = 0;                                                // stride1 hi

  i32x4 g2; g2[0] = 1; g2[1] = 1; g2[2] = 0; g2[3] = 0;     // tensor_dim2=1, dim3=1
  i32x4 g3; g3[0] = 0; g3[1] = (int)(1u << 16); g3[2] = 0; g3[3] = 0; // tensor_dim4=1

#if __clang_major__ >= 23
  i32x8 gx = {0, 0, 0, 0, 0, 0, 0, 0};
  __builtin_amdgcn_tensor_load_to_lds(g0, g1, g2, g3, gx, 0);
#else
  __builtin_amdgcn_tensor_load_to_lds(g0, g1, g2, g3, 0);
#endif
}
#endif

__device__ __forceinline__ void butterfly_store(v4f a, v4f h, v4f v, v4f d,
                                                float* __restrict__ o0) {
  // tl=(A+H+V+D)/2  tr=(A+H-V-D)/2  bl=(A-H+V-D)/2  br=(A-H-V+D)/2
  v4f s0 = a + h, d0 = a - h, s1 = v + d, d1 = v - d;
  v4f tl = (s0 + s1) * 0.5f;
  v4f tr = (s0 - s1) * 0.5f;
  v4f bl = (d0 + d1) * 0.5f;
  v4f br = (d0 - d1) * 0.5f;
  v4f top0 = {tl.x, tr.x, tl.y, tr.y};
  v4f top1 = {tl.z, tr.z, tl.w, tr.w};
  v4f bot0 = {bl.x, br.x, bl.y, br.y};
  v4f bot1 = {bl.z, br.z, bl.w, br.w};
  float* o1 = o0 + OW;
  __builtin_nontemporal_store(top0, (v4f*)o0);
  __builtin_nontemporal_store(top1, (v4f*)(o0 + 4));
  __builtin_nontemporal_store(bot0, (v4f*)o1);
  __builtin_nontemporal_store(bot1, (v4f*)(o1 + 4));
}

__global__ __launch_bounds__(THREADS) void idwt2_haar_kernel(
    const float* __restrict__ coeffs, float* __restrict__ out) {
  const int tid = threadIdx.x;
  const int r0  = blockIdx.x * ROWS_PER_BLK;
  const size_t plane = (size_t)CH * CW;

#if USE_TDM
  // 2 buffers x 4 planes x (ROWS_PER_BLK*CHUNK_W/4) v4f = 64 KB LDS.
  __shared__ v4f stage[2][4][ROWS_PER_BLK * CHUNK_W / 4];

  // The TDM writes LDS behind the compiler's back. Without an escape + may-
  // write, LLVM proves 'stage' has no stores and folds the LDS loads to undef
  // (observed: valu=13, ds=0). Capture the address in an opaque asm with a
  // memory clobber so 'stage' is escaped and potentially modified.
  {
    void* esc = (void*)stage;
    asm volatile("" : "+v"(esc)::"memory");
  }
  // Flat LDS pointers carry the workgroup-relative LDS byte address in their
  // low 32 bits (ISA 10.2) -> descriptor lds_addr.
  const uint32_t lds_base   = (uint32_t)(uintptr_t)(void*)stage;
  const uint32_t PLANE_BYTES = ROWS_PER_BLK * CHUNK_W * 4;  // 8192

  auto issue_chunk = [&](int c, int b) {
    const float* base = coeffs + (size_t)r0 * CW + (size_t)c * CHUNK_W;
#pragma unroll
    for (int p = 0; p < 4; ++p)
      tdm_load_tile(base + (size_t)p * plane,
                    lds_base + (uint32_t)(b * 4 + p) * PLANE_BYTES);
  };

  if (tid < 32) issue_chunk(0, 0);       // wave 0 only; TDM ops are wave-level

#pragma unroll
  for (int c = 0; c < NCHUNK; ++c) {
    const int b = c & 1;
    if (tid < 32) {
      if (c + 1 < NCHUNK) {
        issue_chunk(c + 1, b ^ 1);
        __builtin_amdgcn_s_wait_tensorcnt(4); // chunk c done (TDM is in-order)
      } else {
        __builtin_amdgcn_s_wait_tensorcnt(0);
      }
    }
    __syncthreads();                     // publish chunk c's LDS data to all waves
    asm volatile("" ::: "memory");       // no load CSE across buffer reuse

    const int c0out = 2 * c * CHUNK_W;
#pragma unroll
    for (int r = 0; r < ROWS_PER_BLK; ++r) {
      const int idx = r * (CHUNK_W / 4) + tid;
      v4f a = stage[b][0][idx];
      v4f h = stage[b][1][idx];
      v4f v = stage[b][2][idx];
      v4f d = stage[b][3][idx];
      float* o0 = out + (size_t)(2 * (r0 + r)) * OW + c0out + tid * 8;
      butterfly_store(a, h, v, d, o0);
    }
    __syncthreads();                     // buffer b may be overwritten next iter
  }
#else
  // Fallback: direct streaming with non-temporal b128 loads (read-once data).
  for (int c = 0; c < NCHUNK; ++c) {
    const int c0out = 2 * c * CHUNK_W;
#pragma unroll
    for (int r = 0; r < ROWS_PER_BLK; ++r) {
      const size_t off = (size_t)(r0 + r) * CW + (size_t)c * CHUNK_W + tid * 4;
      v4f a = __builtin_nontemporal_load((const v4f*)(coeffs + off));
      v4f h = __builtin_nontemporal_load((const v4f*)(coeffs + plane + off));
      v4f v = __builtin_nontemporal_load((const v4f*)(coeffs + 2 * plane + off));
      v4f d = __builtin_nontemporal_load((const v4f*)(coeffs + 3 * plane + off));
      float* o0 = out + (size_t)(2 * (r0 + r)) * OW + c0out + tid * 8;
      butterfly_store(a, h, v, d, o0);
    }
  }
#endif
}

extern "C" void kernel_launch(void* const* d_in, const int* in_sizes, int n_in,
                              void* d_out, int out_size, void* d_ws, size_t ws_size,
                              hipStream_t stream) {
  (void)in_sizes; (void)n_in; (void)d_ws; (void)ws_size; (void)out_size;
  const float* coeffs = (const float*)d_in[0];
  float* out = (float*)d_out;
  idwt2_haar_kernel<<<dim3(CH / ROWS_PER_BLK), dim3(THREADS), 0, stream>>>(coeffs, out);
}